// PGcn_31147102830652
// MI455X (gfx1250) — compile-verified
//
#include <hip/hip_runtime.h>

#define EMB_D 128
#define WAVES_PER_BLOCK 8   // blockDim.x == 256

typedef int v4i __attribute__((ext_vector_type(4)));

#if defined(__AMDGCN__) && \
    __has_builtin(__builtin_amdgcn_global_load_async_to_lds_b128) && \
    __has_builtin(__builtin_amdgcn_s_wait_asynccnt)
#define HAVE_ASYNC_LDS 1
#else
#define HAVE_ASYNC_LDS 0
#endif

// --- wave-uniform broadcast via v_readlane_b32 (SGPR result; no ds_bpermute) ---
__device__ __forceinline__ int rdlane_i(int x, int l) {
    return __builtin_amdgcn_readlane(x, l);
}
__device__ __forceinline__ float rdlane_f(float x, int l) {
    return __int_as_float(__builtin_amdgcn_readlane(__float_as_int(x), l));
}

#if HAVE_ASYNC_LDS
// One global_load_async_to_lds_b128 moves a full group's metadata (384B):
// lanes 0-7 -> vals[base..base+31], lanes 8-15 -> rows, lanes 16-23 -> cols
// (16B per lane). Lanes 24-31 are masked off by the branch (partial EXEC).
__device__ __forceinline__ void issue_meta_async(
        const float* vals, const int* rows, const int* cols, int base, int lane,
        __attribute__((address_space(3))) v4i* dstBase) {
    if (lane < 24) {
        const int grp = lane >> 3;   // 0: vals, 1: rows, 2: cols
        const int sub = lane & 7;    // 8 lanes x 16B = 128B per array
        const int* sb = (grp == 0) ? (const int*)vals : (grp == 1) ? rows : cols;
        __attribute__((address_space(1))) v4i* src =
            (__attribute__((address_space(1))) v4i*)(sb + base) + sub;
        __builtin_amdgcn_global_load_async_to_lds_b128(src, dstBase + grp * 8 + sub, 0, 0);
    }
}
#endif

// Fused bidirectional edge-parallel SpMM:
//   y_user[row] += val * x_item[col]
//   y_item[col] += val * x_user[row]
// One wave32 per edge; lane l owns dims [4l, 4l+3] (one B128 per gather).
// Metadata for the NEXT group of 32 edges is staged into LDS by the async
// DMA path one group ahead; source-row gathers are software-pipelined one
// edge ahead in VGPRs with an L2 prefetch two edges ahead.
__global__ void lgcn_edge_spmm(const float* __restrict__ vals,
                               const int*   __restrict__ rows,
                               const int*   __restrict__ cols,
                               int nnz,
                               const float* __restrict__ x_user,
                               const float* __restrict__ x_item,
                               float* __restrict__ y_user,
                               float* __restrict__ y_item) {
    const int lane       = threadIdx.x & 31;
    const int wib        = threadIdx.x >> 5;
    const int wave       = blockIdx.x * WAVES_PER_BLOCK + wib;
    const int totalWaves = gridDim.x * WAVES_PER_BLOCK;
    const int numGroups  = (nnz + 31) >> 5;

#if HAVE_ASYNC_LDS
    __shared__ int metaLds[WAVES_PER_BLOCK][2][96];  // 6KB/block, wave-private slots
    int  buf  = 0;
    bool pend = false;
    if (wave < numGroups && (wave << 5) + 32 <= nnz) {
        issue_meta_async(vals, rows, cols, wave << 5, lane,
                         (__attribute__((address_space(3))) v4i*)&metaLds[wib][0][0]);
        pend = true;
    }
#endif

    for (int g = wave; g < numGroups; g += totalWaves) {
        const int base = g << 5;
        float v;
        int   r, c, cnt;

#if HAVE_ASYNC_LDS
        if (pend) {
            __builtin_amdgcn_s_wait_asynccnt(0);
            v   = __int_as_float(metaLds[wib][buf][lane]);
            r   = metaLds[wib][buf][32 + lane];
            c   = metaLds[wib][buf][64 + lane];
            cnt = 32;
        } else
#endif
        {
            const int e = base + lane;
            v = 0.0f; r = 0; c = 0;
            if (e < nnz) { v = vals[e]; r = rows[e]; c = cols[e]; }
            cnt = nnz - base;
            if (cnt > 32) cnt = 32;
        }

#if HAVE_ASYNC_LDS
        {   // Stage next group's metadata into the other buffer (<=1 in flight).
            const int gn = g + totalWaves;
            const int nb = buf ^ 1;
            if (gn < numGroups && (gn << 5) + 32 <= nnz) {
                issue_meta_async(vals, rows, cols, gn << 5, lane,
                                 (__attribute__((address_space(3))) v4i*)&metaLds[wib][nb][0]);
                pend = true;
            } else {
                pend = false;
            }
            buf = nb;
        }
#endif

        // Pipeline prologue: stage edge 0's source rows.
        int   rj = rdlane_i(r, 0);
        int   cj = rdlane_i(c, 0);
        float vj = rdlane_f(v, 0);
        float4 xi = ((const float4*)(x_item + (size_t)cj * EMB_D))[lane];
        float4 xu = ((const float4*)(x_user + (size_t)rj * EMB_D))[lane];

        for (int j = 0; j < cnt; ++j) {
            const int    rcur = rj;
            const int    ccur = cj;
            const float  vcur = vj;
            const float4 xic  = xi;
            const float4 xuc  = xu;

            // Stage edge j+1 (loads issue before the atomic clause below).
            if (j + 1 < cnt) {
                rj = rdlane_i(r, j + 1);
                cj = rdlane_i(c, j + 1);
                vj = rdlane_f(v, j + 1);
                xi = ((const float4*)(x_item + (size_t)cj * EMB_D))[lane];
                xu = ((const float4*)(x_user + (size_t)rj * EMB_D))[lane];
            }
            // L2 prefetch for edge j+2 (global_prefetch_b8).
            if (j + 2 < cnt) {
                const int r2 = rdlane_i(r, j + 2);
                const int c2 = rdlane_i(c, j + 2);
                __builtin_prefetch(x_item + (size_t)c2 * EMB_D + lane * 4, 0, 0);
                __builtin_prefetch(x_user + (size_t)r2 * EMB_D + lane * 4, 0, 0);
            }

            float* yu = y_user + (size_t)rcur * EMB_D + lane * 4;
            float* yi = y_item + (size_t)ccur * EMB_D + lane * 4;

            // Native non-returning global_atomic_add_f32; targets L2-resident.
            unsafeAtomicAdd(yu + 0, vcur * xic.x);
            unsafeAtomicAdd(yu + 1, vcur * xic.y);
            unsafeAtomicAdd(yu + 2, vcur * xic.z);
            unsafeAtomicAdd(yu + 3, vcur * xic.w);

            unsafeAtomicAdd(yi + 0, vcur * xuc.x);
            unsafeAtomicAdd(yi + 1, vcur * xuc.y);
            unsafeAtomicAdd(yi + 2, vcur * xuc.z);
            unsafeAtomicAdd(yi + 3, vcur * xuc.w);
        }
    }
}

// out = (acc + next) * scale, float4-vectorized (B128 load/store).
// acc may equal out (in-place accumulate) or be a different source
// (fused init on the first layer, avoiding a separate d2d memcpy pass).
__global__ void lgcn_acc_scale(float* __restrict__ out,
                               const float* __restrict__ acc,
                               const float* __restrict__ next,
                               int n4, float scale) {
    int idx    = blockIdx.x * blockDim.x + threadIdx.x;
    int stride = gridDim.x * blockDim.x;
    float4*       o4  = (float4*)out;
    const float4* a4  = (const float4*)acc;
    const float4* n4p = (const float4*)next;
    for (int i = idx; i < n4; i += stride) {
        float4 a = a4[i];
        float4 x = n4p[i];
        a.x = (a.x + x.x) * scale;
        a.y = (a.y + x.y) * scale;
        a.z = (a.z + x.z) * scale;
        a.w = (a.w + x.w) * scale;
        o4[i] = a;
    }
}

extern "C" void kernel_launch(void* const* d_in, const int* in_sizes, int n_in,
                              void* d_out, int out_size, void* d_ws, size_t ws_size,
                              hipStream_t stream) {
    const float* user_pref = (const float*)d_in[0];
    const float* item_pref = (const float*)d_in[1];
    const float* vals      = (const float*)d_in[2];
    const int*   rows      = (const int*)d_in[3];
    const int*   cols      = (const int*)d_in[4];

    const int U   = in_sizes[0] / EMB_D;   // 100000
    const int I   = in_sizes[1] / EMB_D;   // 50000
    const int nnz = in_sizes[2];           // 2000000

    const size_t uElems = (size_t)U * EMB_D;
    const size_t iElems = (size_t)I * EMB_D;

    float* out_u = (float*)d_out;
    float* out_i = out_u + uElems;

    // Workspace: two ping-pong layer buffers per side (153.6 MB total).
    float* ws = (float*)d_ws;
    float* uA = ws;
    float* uB = uA + uElems;
    float* iA = uB + uElems;
    float* iB = iA + iElems;

    const float* cur_u = user_pref;
    const float* cur_i = item_pref;

    const int n4u = (int)(uElems / 4);
    const int n4i = (int)(iElems / 4);

    for (int layer = 0; layer < 3; ++layer) {
        float* nu = (layer & 1) ? uB : uA;
        float* ni = (layer & 1) ? iB : iA;

        (void)hipMemsetAsync(nu, 0, uElems * sizeof(float), stream);
        (void)hipMemsetAsync(ni, 0, iElems * sizeof(float), stream);

        lgcn_edge_spmm<<<4096, 256, 0, stream>>>(vals, rows, cols, nnz,
                                                 cur_u, cur_i, nu, ni);

        // Layer 0 reads the accumulator source directly from the inputs
        // (fused init, no memcpy pass); later layers accumulate in place.
        const float* acc_u = (layer == 0) ? user_pref : out_u;
        const float* acc_i = (layer == 0) ? item_pref : out_i;
        const float s = (layer == 2) ? 0.25f : 1.0f;  // fold 1/(L+1) into last add

        lgcn_acc_scale<<<(n4u + 255) / 256, 256, 0, stream>>>(out_u, acc_u, nu, n4u, s);
        lgcn_acc_scale<<<(n4i + 255) / 256, 256, 0, stream>>>(out_i, acc_i, ni, n4i, s);

        cur_u = nu;
        cur_i = ni;
    }
}